// MKLGIN_26087631356380
// MI455X (gfx1250) — compile-verified
//
#include <hip/hip_runtime.h>

#define N_NODES 100000
#define N_EDGES 1600000
#define D_IN 64
#define D_HID 256
#define LDSS 260   // padded LDS stride (260 % 64 == 4) -> conflict-free K-major reads

typedef __attribute__((ext_vector_type(2))) float v2f;
typedef __attribute__((ext_vector_type(8))) float v8f;

// ---- Phase 1: y = (1 + eps) * x -------------------------------------------
__global__ void gin_scale_kernel(const float* __restrict__ x,
                                 const float* __restrict__ eps,
                                 float* __restrict__ y, int n) {
  int i = blockIdx.x * blockDim.x + threadIdx.x;
  if (i < n) y[i] = (1.0f + eps[0]) * x[i];
}

// ---- Phase 2: y[dst] += x[src] over all edges (native f32 atomics) --------
__global__ void gin_scatter_kernel(const float* __restrict__ x,
                                   const long long* __restrict__ ei,
                                   float* __restrict__ y) {
  int tid = blockIdx.x * blockDim.x + threadIdx.x;
  int e = tid >> 4;           // edge id
  int c = (tid & 15) << 2;    // feature chunk (float4), coalesced across 16 lanes
  if (e >= N_EDGES) return;
  long long src = ei[e];
  long long dst = ei[(long long)N_EDGES + e];
  const float4 v = *(const float4*)(x + src * D_IN + c);
  float* o = y + dst * D_IN + c;
  unsafeAtomicAdd(o + 0, v.x);   // -> global_atomic_add_f32
  unsafeAtomicAdd(o + 1, v.y);
  unsafeAtomicAdd(o + 2, v.z);
  unsafeAtomicAdd(o + 3, v.w);
}

// ---- Phase 3: out = relu(y@W1 + b1) @ W2 + b2, one 16-row tile per block --
// 4 waves / block. GEMM1: each wave owns 64 of 256 N-columns (4 x 16x16 tiles).
// GEMM2: each wave owns 16 of 64 N-columns (1 x 16x16 tile).
// Full-precision f32 WMMA: V_WMMA_F32_16X16X4_F32.
__global__ void __launch_bounds__(128)
gin_mlp_kernel(const float* __restrict__ y,
               const float* __restrict__ W1, const float* __restrict__ b1,
               const float* __restrict__ W2, const float* __restrict__ b2,
               float* __restrict__ out) {
  __shared__ float h_lds[16 * LDSS];            // 16x256 hidden tile, padded

  const int lane = threadIdx.x & 31;
  const int wave = threadIdx.x >> 5;            // 0..3
  const int m    = lane & 15;                   // M row (A) / N col (B,C)
  const int kh   = (lane >> 4) << 1;            // K sub-offset: 0 or 2
  const int mrow = (lane >> 4) << 3;            // C/D row offset: 0 or 8
  const long long row0 = (long long)blockIdx.x * 16;

  // ---- GEMM1: h = relu(y_tile[16x64] @ W1[64x256] + b1) ----
  // Preload all 16 A fragments (K = 64, 4 per WMMA step); reused for 4 N-tiles.
  v2f afrag[16];
  const float* yrow = y + (row0 + m) * D_IN + kh;
#pragma unroll
  for (int ks = 0; ks < 16; ++ks)
    afrag[ks] = *(const v2f*)(yrow + ks * 4);

#pragma unroll
  for (int nt = 0; nt < 4; ++nt) {
    const int n0 = wave * 64 + nt * 16;
    v8f acc = {};
#pragma unroll
    for (int ks = 0; ks < 16; ++ks) {
      const int k = ks * 4 + kh;
      v2f bfrag;
      bfrag.x = W1[(k + 0) * D_HID + n0 + m];
      bfrag.y = W1[(k + 1) * D_HID + n0 + m];
      acc = __builtin_amdgcn_wmma_f32_16x16x4_f32(false, afrag[ks], false, bfrag,
                                                  (short)0, acc, false, false);
    }
    const float bias = b1[n0 + m];
#pragma unroll
    for (int r = 0; r < 8; ++r) {
      float h = acc[r] + bias;
      h_lds[(r + mrow) * LDSS + n0 + m] = h > 0.0f ? h : 0.0f;
    }
  }
  __syncthreads();

  // ---- GEMM2: out_tile = h[16x256] @ W2[256x64] + b2 ----
  {
    const int n0 = wave * 16;
    v8f acc = {};
#pragma unroll
    for (int ks = 0; ks < 64; ++ks) {
      const int k = ks * 4 + kh;
      v2f af = *(const v2f*)(&h_lds[m * LDSS + k]);   // conflict-free (stride 260)
      v2f bf;
      bf.x = W2[(k + 0) * D_IN + n0 + m];
      bf.y = W2[(k + 1) * D_IN + n0 + m];
      acc = __builtin_amdgcn_wmma_f32_16x16x4_f32(false, af, false, bf,
                                                  (short)0, acc, false, false);
    }
    const float bias = b2[n0 + m];
    float* orow = out + (row0 + mrow) * D_IN + n0 + m;
#pragma unroll
    for (int r = 0; r < 8; ++r)
      orow[r * D_IN] = acc[r] + bias;
  }
}

extern "C" void kernel_launch(void* const* d_in, const int* in_sizes, int n_in,
                              void* d_out, int out_size, void* d_ws, size_t ws_size,
                              hipStream_t stream) {
  const float*      x   = (const float*)d_in[0];
  const long long*  ei  = (const long long*)d_in[1];
  const float*      eps = (const float*)d_in[2];
  const float*      W1  = (const float*)d_in[3];
  const float*      b1  = (const float*)d_in[4];
  const float*      W2  = (const float*)d_in[5];
  const float*      b2  = (const float*)d_in[6];
  float* out = (float*)d_out;
  float* y   = (float*)d_ws;                  // N_NODES * D_IN floats (25.6 MB)

  const int n = N_NODES * D_IN;
  gin_scale_kernel<<<(n + 255) / 256, 256, 0, stream>>>(x, eps, y, n);

  const int nscat = N_EDGES * 16;             // one thread per edge x float4 chunk
  gin_scatter_kernel<<<nscat / 256, 256, 0, stream>>>(x, ei, y);

  gin_mlp_kernel<<<N_NODES / 16, 128, 0, stream>>>(y, W1, b1, W2, b2, out);
}